// Model_48850958025309
// MI455X (gfx1250) — compile-verified
//
#include <hip/hip_runtime.h>
#include <math.h>

// ---- problem sizes ----
#define L       336
#define P       96
#define NCH     300
#define BATCH   8
#define BN      2400      // BATCH*NCH
#define KD      672       // 2*L (v and v^2 concatenated)
#define C2      192       // 2*P (two taylor experts)
#define EC      384       // 4*P expert-out columns
#define MH_C    0.8673250705840776f

typedef float v2f __attribute__((ext_vector_type(2)));
typedef float v8f __attribute__((ext_vector_type(8)));

__device__ __forceinline__ float wred(float v) {
  #pragma unroll
  for (int off = 16; off > 0; off >>= 1) v += __shfl_xor(v, off, 32);
  return v;
}

// ---------------- RevIN: stats + v, v^2, v^T ----------------
__global__ void k_revin(const float* __restrict__ x, float* __restrict__ VCAT,
                        float* __restrict__ VT, float* __restrict__ MEANV,
                        float* __restrict__ STDV) {
  int wave = (blockIdx.x * blockDim.x + threadIdx.x) >> 5;
  int lane = threadIdx.x & 31;
  if (wave >= BN / 32) return;
  int r = wave * 32 + lane;              // row = b*NCH + n
  int b = r / NCH, n = r % NCH;
  const float* xp = x + (size_t)b * L * NCH + n;
  float s = 0.f, ss = 0.f;
  for (int l = 0; l < L; ++l) { float v = xp[(size_t)l * NCH]; s += v; ss += v * v; }
  float mean  = s * (1.0f / L);
  float var   = ss * (1.0f / L) - mean * mean;
  float stdev = sqrtf(var + 1e-5f);
  MEANV[r] = mean; STDV[r] = stdev;
  for (int l = 0; l < L; ++l) {
    float v = (xp[(size_t)l * NCH] - mean) / stdev;
    VCAT[(size_t)r * KD + l]     = v;
    VCAT[(size_t)r * KD + L + l] = v * v;
    VT[(size_t)l * BN + r]       = v;          // coalesced layout for wave-KAN
  }
}

// ---- prep: B^T [C2][KD] for taylor GEMM, const col, reciprocal scales ----
__global__ void k_prep(const float* __restrict__ t1c, const float* __restrict__ t1b,
                       const float* __restrict__ t2c, const float* __restrict__ t2b,
                       const float* __restrict__ w1s, const float* __restrict__ w2s,
                       float* __restrict__ BMT, float* __restrict__ CONSTC,
                       float* __restrict__ RCP) {
  int wave = (blockIdx.x * blockDim.x + threadIdx.x) >> 5;
  int lane = threadIdx.x & 31;
  if (wave >= C2) return;
  int col = wave, e = col / P, o = col % P;
  const float* cf = e ? t2c : t1c;
  const float* bs = e ? t2b : t1b;
  const float* sc = e ? w2s : w1s;
  float c0 = 0.f;
  for (int i = lane; i < L; i += 32) {
    size_t base = ((size_t)o * L + i) * 3;
    c0 += cf[base + 0];
    BMT[(size_t)col * KD + i]     = cf[base + 1];
    BMT[(size_t)col * KD + L + i] = cf[base + 2];
    RCP[(size_t)e * P * L + (size_t)o * L + i] = 1.0f / sc[(size_t)o * L + i];
  }
  c0 = wred(c0);
  if (lane == 0) CONSTC[col] = c0 + bs[o];
}

// ---------------- gate: softmax(v @ gate_w^T + b) ----------------
__global__ void k_gate(const float* __restrict__ VCAT, const float* __restrict__ gw,
                       const float* __restrict__ gb, float* __restrict__ SCORE) {
  int wave = (blockIdx.x * blockDim.x + threadIdx.x) >> 5;
  int lane = threadIdx.x & 31;
  if (wave >= BN) return;
  int r = wave;
  float d0 = 0, d1 = 0, d2 = 0, d3 = 0;
  for (int i = lane; i < L; i += 32) {
    float v = VCAT[(size_t)r * KD + i];
    d0 += v * gw[i]; d1 += v * gw[L + i]; d2 += v * gw[2 * L + i]; d3 += v * gw[3 * L + i];
  }
  d0 = wred(d0) + gb[0]; d1 = wred(d1) + gb[1];
  d2 = wred(d2) + gb[2]; d3 = wred(d3) + gb[3];
  float m  = fmaxf(fmaxf(d0, d1), fmaxf(d2, d3));
  float e0 = expf(d0 - m), e1 = expf(d1 - m), e2 = expf(d2 - m), e3 = expf(d3 - m);
  float z  = e0 + e1 + e2 + e3;
  if (lane == 0) {
    SCORE[r * 4 + 0] = e0 / z; SCORE[r * 4 + 1] = e1 / z;
    SCORE[r * 4 + 2] = e2 / z; SCORE[r * 4 + 3] = e3 / z;
  }
}

// ------- taylor experts: [BN x KD] @ [KD x C2] via V_WMMA_F32_16X16X4_F32 -------
__global__ void k_taylor_wmma(const float* __restrict__ VCAT, const float* __restrict__ BMT,
                              const float* __restrict__ CONSTC, float* __restrict__ EXPO) {
  int wave = (blockIdx.x * blockDim.x + threadIdx.x) >> 5;
  int lane = threadIdx.x & 31;
  int tm = wave / (C2 / 16);             // 0..149
  int tn = wave % (C2 / 16);             // 0..11
  if (tm >= BN / 16) return;             // wave-uniform; EXEC stays all-ones
  int half = lane >> 4;                  // K-half select (A: K 0-1 vs 2-3)
  int lm   = lane & 15;
  const float* arow = VCAT + (size_t)(tm * 16 + lm) * KD + half * 2;
  const float* brow = BMT  + (size_t)(tn * 16 + lm) * KD + half * 2;
  v8f acc = {};
  for (int k0 = 0; k0 < KD; k0 += 4) {
    v2f a  = *(const v2f*)(arow + k0);
    v2f bb = *(const v2f*)(brow + k0);
    acc = __builtin_amdgcn_wmma_f32_16x16x4_f32(false, a, false, bb,
                                                (short)0, acc, false, false);
  }
  int rowbase = tm * 16 + half * 8;      // D: VGPR r -> M = r + 8*(lane>=16)
  int col     = tn * 16 + lm;            // N = lane&15
  float cadd  = CONSTC[col];
  #pragma unroll
  for (int rr = 0; rr < 8; ++rr)
    EXPO[(size_t)(rowbase + rr) * EC + col] = acc[rr] + cadd;
}

// ------- wave experts: mexican-hat KAN, raw (pre-BN) outputs -------
__global__ void k_wave(const float* __restrict__ VT,
                       const float* __restrict__ w1t, const float* __restrict__ w1w,
                       const float* __restrict__ w2t, const float* __restrict__ w2w,
                       const float* __restrict__ RCP, float* __restrict__ EXPO) {
  int wave = (blockIdx.x * blockDim.x + threadIdx.x) >> 5;
  int lane = threadIdx.x & 31;
  int b32  = wave % (BN / 32);
  int o    = (wave / (BN / 32)) % P;
  int e    = wave / ((BN / 32) * P);     // 0 or 1
  int r    = b32 * 32 + lane;
  const float* tp = (e ? w2t : w1t) + (size_t)o * L;
  const float* wp = (e ? w2w : w1w) + (size_t)o * L;
  const float* rp = RCP + (size_t)e * P * L + (size_t)o * L;
  float acc = 0.f;
  for (int i = 0; i < L; ++i) {
    float xv  = VT[(size_t)i * BN + r];          // coalesced
    float xs  = (xv - tp[i]) * rp[i];
    float x2  = xs * xs;
    float psi = (x2 - 1.0f) * __expf(-0.5f * x2);  // v_exp_f32
    acc = fmaf(psi, wp[i], acc);
  }
  EXPO[(size_t)r * EC + 2 * P + e * P + o] = MH_C * acc;
}

// ------- batchnorm column stats for wave experts: fold into A*x+B -------
__global__ void k_stats(const float* __restrict__ EXPO,
                        const float* __restrict__ g1, const float* __restrict__ b1,
                        const float* __restrict__ g2, const float* __restrict__ b2,
                        float* __restrict__ ACO, float* __restrict__ BCO) {
  int wave = (blockIdx.x * blockDim.x + threadIdx.x) >> 5;
  int lane = threadIdx.x & 31;
  if (wave >= C2) return;
  int col = wave;
  float s = 0.f, ss = 0.f;
  for (int r = lane; r < BN; r += 32) {
    float v = EXPO[(size_t)r * EC + 2 * P + col];
    s += v; ss += v * v;
  }
  s = wred(s); ss = wred(ss);
  float mu   = s * (1.0f / BN);
  float var  = ss * (1.0f / BN) - mu * mu;
  float rstd = 1.0f / sqrtf(var + 1e-5f);
  float gamma = (col < P) ? g1[col] : g2[col - P];
  float beta  = (col < P) ? b1[col] : b2[col - P];
  float a = rstd * gamma;
  if (lane == 0) { ACO[col] = a; BCO[col] = beta - mu * a; }
}

// ------- combine experts, denorm, classifier -------
__global__ void k_combine(const float* __restrict__ EXPO, const float* __restrict__ SCORE,
                          const float* __restrict__ ACO, const float* __restrict__ BCO,
                          const float* __restrict__ MEANV, const float* __restrict__ STDV,
                          const float* __restrict__ clsw, const float* __restrict__ clsb,
                          float* __restrict__ out) {
  int wave = (blockIdx.x * blockDim.x + threadIdx.x) >> 5;
  int lane = threadIdx.x & 31;
  if (wave >= BATCH * P) return;
  int b = wave / P, p = wave % P;
  float A1 = ACO[p], B1 = BCO[p], A2 = ACO[P + p], B2 = BCO[P + p];
  float acc0 = 0.f, acc1 = 0.f;
  for (int it = 0; it < (NCH + 31) / 32; ++it) {
    int n = it * 32 + lane;
    if (n < NCH) {
      int r = b * NCH + n;
      const float* er = EXPO + (size_t)r * EC;
      const float* sr = SCORE + (size_t)r * 4;
      float t1v = er[p],            t2v = er[P + p];
      float w1v = er[2 * P + p] * A1 + B1;
      float w2v = er[3 * P + p] * A2 + B2;
      float pred = sr[0] * t1v + sr[1] * t2v + sr[2] * w1v + sr[3] * w2v;
      float pd   = pred * STDV[r] + MEANV[r];
      acc0 += pd * clsw[n];
      acc1 += pd * clsw[NCH + n];
    }
  }
  acc0 = wred(acc0); acc1 = wred(acc1);
  if (lane == 0) {
    out[((size_t)b * P + p) * 2 + 0] = acc0 + clsb[0];
    out[((size_t)b * P + p) * 2 + 1] = acc1 + clsb[1];
  }
}

extern "C" void kernel_launch(void* const* d_in, const int* in_sizes, int n_in,
                              void* d_out, int out_size, void* d_ws, size_t ws_size,
                              hipStream_t stream) {
  const float* x     = (const float*)d_in[0];
  const float* gw    = (const float*)d_in[1];
  const float* gb    = (const float*)d_in[2];
  const float* t1c   = (const float*)d_in[3];
  const float* t1b   = (const float*)d_in[4];
  const float* t2c   = (const float*)d_in[5];
  const float* t2b   = (const float*)d_in[6];
  const float* w1s   = (const float*)d_in[7];
  const float* w1t   = (const float*)d_in[8];
  const float* w1w   = (const float*)d_in[9];
  const float* w1g   = (const float*)d_in[10];
  const float* w1b   = (const float*)d_in[11];
  const float* w2s   = (const float*)d_in[12];
  const float* w2t   = (const float*)d_in[13];
  const float* w2w   = (const float*)d_in[14];
  const float* w2g   = (const float*)d_in[15];
  const float* w2b   = (const float*)d_in[16];
  const float* clsw  = (const float*)d_in[17];
  const float* clsb  = (const float*)d_in[18];
  float* out = (float*)d_out;

  char* ws = (char*)d_ws;
  size_t off = 0;
  auto alloc = [&](size_t bytes) -> float* {
    float* p = (float*)(ws + off);
    off += (bytes + 255) & ~(size_t)255;
    return p;
  };
  float* VCAT   = alloc((size_t)BN * KD * 4);   // [2400][672]
  float* VT     = alloc((size_t)L * BN * 4);    // [336][2400]
  float* MEANV  = alloc((size_t)BN * 4);
  float* STDV   = alloc((size_t)BN * 4);
  float* BMT    = alloc((size_t)C2 * KD * 4);   // [192][672]
  float* CONSTC = alloc((size_t)C2 * 4);
  float* RCP    = alloc((size_t)2 * P * L * 4);
  float* EXPO   = alloc((size_t)BN * EC * 4);   // [2400][384]
  float* SCORE  = alloc((size_t)BN * 4 * 4);
  float* ACO    = alloc((size_t)C2 * 4);
  float* BCO    = alloc((size_t)C2 * 4);

  const dim3 B256(256);
  k_revin<<<dim3((BN / 32 * 32 + 255) / 256), B256, 0, stream>>>(x, VCAT, VT, MEANV, STDV);
  k_prep<<<dim3(C2 * 32 / 256), B256, 0, stream>>>(t1c, t1b, t2c, t2b, w1s, w2s,
                                                   BMT, CONSTC, RCP);
  k_gate<<<dim3(BN * 32 / 256), B256, 0, stream>>>(VCAT, gw, gb, SCORE);
  k_taylor_wmma<<<dim3((BN / 16) * (C2 / 16) * 32 / 256), B256, 0, stream>>>(VCAT, BMT,
                                                                             CONSTC, EXPO);
  k_wave<<<dim3(2 * P * (BN / 32) * 32 / 256), B256, 0, stream>>>(VT, w1t, w1w, w2t, w2w,
                                                                  RCP, EXPO);
  k_stats<<<dim3(C2 * 32 / 256), B256, 0, stream>>>(EXPO, w1g, w1b, w2g, w2b, ACO, BCO);
  k_combine<<<dim3(BATCH * P * 32 / 256), B256, 0, stream>>>(EXPO, SCORE, ACO, BCO,
                                                             MEANV, STDV, clsw, clsb, out);
}